// TitansMemoryModule_19524921327968
// MI455X (gfx1250) — compile-verified
//
#include <hip/hip_runtime.h>
#include <hip/hip_bf16.h>

// ---------------- problem constants ----------------
constexpr int H  = 16;
constexpr int L  = 2048;
constexpr int D  = 64;
constexpr int CK = 64;          // chunk (tokens)
constexpr int NC = L / CK;      // 32 chunks
constexpr int PITCH = D + 4;    // LDS pitch (floats); 272B rows keep 16B alignment
#define EPSLN 1e-6f

typedef float v2f __attribute__((ext_vector_type(2)));
typedef float v8f __attribute__((ext_vector_type(8)));

__device__ __forceinline__ v8f wmma_f32_16x16x4(v2f a, v2f b, v8f c) {
  // D = A(16x4 f32) * B(4x16 f32) + C(16x16 f32)
  return __builtin_amdgcn_wmma_f32_16x16x4_f32(false, a, false, b, (short)0, c,
                                               false, false);
}

__device__ __forceinline__ float wave_sum32(float x) {
#pragma unroll
  for (int off = 16; off > 0; off >>= 1) x += __shfl_xor(x, off, 32);
  return x;
}

// ---- CDNA5 async global->LDS copy (ASYNCcnt path), 16B per lane ----
__device__ __forceinline__ void async_copy16(const float* gsrc, float* ldst) {
  unsigned ldsoff = (unsigned)(uintptr_t)ldst;  // low 32 bits = LDS byte offset
  asm volatile("global_load_async_to_lds_b128 %0, %1, off"
               :
               : "v"(ldsoff), "v"(gsrc)
               : "memory");
}
__device__ __forceinline__ void wait_async0() {
  asm volatile("s_wait_asynccnt 0" ::: "memory");
}

// A fragment (M x K tile, row-major LDS [m][k]):
// lane l: m = m0 + (l&15); k pair base = k0 + 2*(l>>4)
__device__ __forceinline__ v2f load_a_rm(const float* s, int m0, int k0, int lane) {
  int m = m0 + (lane & 15);
  int kk = k0 + ((lane >> 4) << 1);
  v2f a;
  a[0] = s[m * PITCH + kk + 0];
  a[1] = s[m * PITCH + kk + 1];
  return a;
}
// A fragment from transposed storage (LDS holds [k][m], want A[m][k])
__device__ __forceinline__ v2f load_a_tr(const float* s, int m0, int k0, int lane) {
  int m = m0 + (lane & 15);
  int kk = k0 + ((lane >> 4) << 1);
  v2f a;
  a[0] = s[(kk + 0) * PITCH + m];
  a[1] = s[(kk + 1) * PITCH + m];
  return a;
}
// B fragment (K x N tile, row-major LDS [k][n])
__device__ __forceinline__ v2f load_b_rm(const float* s, int k0, int n0, int lane) {
  int n = n0 + (lane & 15);
  int kk = k0 + ((lane >> 4) << 1);
  v2f b;
  b[0] = s[(kk + 0) * PITCH + n];
  b[1] = s[(kk + 1) * PITCH + n];
  return b;
}
// B fragment where logical B[k][n] is stored as LDS [n][k] (i.e. B = Xᵀ)
__device__ __forceinline__ v2f load_b_tr(const float* s, int k0, int n0, int lane) {
  int n = n0 + (lane & 15);
  int kk = k0 + ((lane >> 4) << 1);
  v2f b;
  b[0] = s[n * PITCH + kk + 0];
  b[1] = s[n * PITCH + kk + 1];
  return b;
}
// B fragment from global row-major [k][n] with row stride D
__device__ __forceinline__ v2f load_b_glb(const float* g, int k0, int n0, int lane) {
  int n = n0 + (lane & 15);
  int kk = k0 + ((lane >> 4) << 1);
  v2f b;
  b[0] = g[(kk + 0) * D + n];
  b[1] = g[(kk + 1) * D + n];
  return b;
}

// =====================================================================
// Kernel 1: per (head, chunk): grad = LNfusedL2Bwd(K@W0, K, V),
//           cp (in-chunk cumprod of gate), P_c = cp[CK-1],
//           U_c = Kᵀ @ diag(-(P_c/cp_j)*eta_j) Grad
// =====================================================================
__global__ __launch_bounds__(256) void ttt_k1_grad(
    const float* __restrict__ kin, const float* __restrict__ vin,
    const float* __restrict__ gate, const float* __restrict__ eta,
    const float* __restrict__ W0, const float* __restrict__ gamma,
    const float* __restrict__ beta, float* __restrict__ grad_ws,
    float* __restrict__ U_ws, float* __restrict__ cp_ws,
    float* __restrict__ Pc_ws) {
  __shared__ float Kls[CK * PITCH];
  __shared__ float Wls[D * PITCH];
  __shared__ float Zls[CK * PITCH];
  __shared__ float cp_s[CK];
  __shared__ float eta_s[CK];

  const int blk = blockIdx.x;
  const int h = blk / NC, c = blk % NC;
  const int t0 = c * CK;
  const int tid = threadIdx.x;
  const int lane = tid & 31, wv = tid >> 5;

  // async-stage K chunk and W0[h] (64x64 each) into LDS, 16B per lane
  for (int i = tid; i < CK * D / 4; i += 256) {
    int r = i >> 4, cc = (i & 15) << 2;
    async_copy16(&kin[((size_t)(h * L + t0 + r)) * D + cc], &Kls[r * PITCH + cc]);
    async_copy16(&W0[((size_t)(h * D + r)) * D + cc], &Wls[r * PITCH + cc]);
  }
  if (tid < CK) eta_s[tid] = eta[h * L + t0 + tid];
  if (tid == 0) {
    float p = 1.0f;
    for (int i = 0; i < CK; ++i) { p *= gate[h * L + t0 + i]; cp_s[i] = p; }
  }
  wait_async0();
  __syncthreads();
  if (tid < CK) cp_ws[h * L + t0 + tid] = cp_s[tid];
  if (tid == 0) Pc_ws[h * NC + c] = cp_s[CK - 1];

  // ---- Z1 = K @ W0 : wave wv owns tiles (ti, tjb..tjb+1) ----
  const int ti = wv & 3, tjb = (wv >> 2) << 1;
#pragma unroll
  for (int tjx = 0; tjx < 2; ++tjx) {
    const int tj = tjb + tjx;
    v8f acc = {};
#pragma unroll
    for (int kb = 0; kb < D / 4; ++kb) {
      v2f a = load_a_rm(Kls, ti * 16, kb * 4, lane);
      v2f b = load_b_rm(Wls, kb * 4, tj * 16, lane);
      acc = wmma_f32_16x16x4(a, b, acc);
    }
#pragma unroll
    for (int e = 0; e < 8; ++e) {
      int r = ti * 16 + e + ((lane >> 4) << 3);
      Zls[r * PITCH + tj * 16 + (lane & 15)] = acc[e];
    }
  }
  __syncthreads();

  // ---- fused LN + L2 backward, row-wise; each wave handles 8 rows ----
  const float cpEnd = cp_s[CK - 1];
  const float g0 = gamma[h * D + lane], g1 = gamma[h * D + lane + 32];
  const float b0 = beta[h * D + lane],  b1 = beta[h * D + lane + 32];
#pragma unroll 1
  for (int rr = 0; rr < 8; ++rr) {
    const int r = wv * 8 + rr;
    float x0 = Zls[r * PITCH + lane], x1 = Zls[r * PITCH + lane + 32];
    float mu = wave_sum32(x0 + x1) * (1.0f / 64.0f);
    float d0 = x0 - mu, d1 = x1 - mu;
    float var = wave_sum32(d0 * d0 + d1 * d1) * (1.0f / 64.0f);
    float rstd = __frsqrt_rn(var + EPSLN);
    float xh0 = d0 * rstd, xh1 = d1 * rstd;
    float kr0 = Kls[r * PITCH + lane], kr1 = Kls[r * PITCH + lane + 32];
    float pv0 = vin[((size_t)(h * L + t0 + r)) * D + lane];
    float pv1 = vin[((size_t)(h * L + t0 + r)) * D + lane + 32];
    float go0 = 2.0f * (g0 * xh0 + b0 + kr0 - pv0);
    float go1 = 2.0f * (g1 * xh1 + b1 + kr1 - pv1);
    float gx0 = go0 * g0, gx1 = go1 * g1;
    float mg  = wave_sum32(gx0 + gx1) * (1.0f / 64.0f);
    float mgx = wave_sum32(gx0 * xh0 + gx1 * xh1) * (1.0f / 64.0f);
    float z0 = (gx0 - mg - xh0 * mgx) * rstd;
    float z1 = (gx1 - mg - xh1 * mgx) * rstd;
    grad_ws[((size_t)(h * L + t0 + r)) * D + lane] = z0;
    grad_ws[((size_t)(h * L + t0 + r)) * D + lane + 32] = z1;
    float coef = -(cpEnd / cp_s[r]) * eta_s[r];
    Zls[r * PITCH + lane] = coef * z0;
    Zls[r * PITCH + lane + 32] = coef * z1;
  }
  __syncthreads();

  // ---- U = Kᵀ @ Gs : A[m=d][k=j] = Kls[j][d] (transposed read) ----
  float* Ubase = U_ws + ((size_t)(h * NC + c)) * D * D;
#pragma unroll
  for (int tjx = 0; tjx < 2; ++tjx) {
    const int tj = tjb + tjx;
    v8f acc = {};
#pragma unroll
    for (int kb = 0; kb < D / 4; ++kb) {
      v2f a = load_a_tr(Kls, ti * 16, kb * 4, lane);
      v2f b = load_b_rm(Zls, kb * 4, tj * 16, lane);
      acc = wmma_f32_16x16x4(a, b, acc);
    }
#pragma unroll
    for (int e = 0; e < 8; ++e) {
      int d = ti * 16 + e + ((lane >> 4) << 3);
      Ubase[d * D + tj * 16 + (lane & 15)] = acc[e];
    }
  }
}

// =====================================================================
// Kernel 2: per head, scan chunk summaries: store W_base[c], then
//           W = P_c * W + U_c   (32 steps, elementwise 64x64)
// =====================================================================
__global__ __launch_bounds__(256) void ttt_k2_scan(
    const float* __restrict__ W0, const float* __restrict__ U_ws,
    const float* __restrict__ Pc_ws, float* __restrict__ Wb_ws) {
  const int h = blockIdx.x;
  const int tid = threadIdx.x;
  float4 w[4];
  const float4* w0p = (const float4*)&W0[(size_t)h * D * D + tid * 16];
#pragma unroll
  for (int i = 0; i < 4; ++i) w[i] = w0p[i];
#pragma unroll 1
  for (int c = 0; c < NC; ++c) {
    float4* wb = (float4*)&Wb_ws[((size_t)(h * NC + c)) * D * D + tid * 16];
    const float4* up = (const float4*)&U_ws[((size_t)(h * NC + c)) * D * D + tid * 16];
    const float p = Pc_ws[h * NC + c];
#pragma unroll
    for (int i = 0; i < 4; ++i) {
      wb[i] = w[i];
      float4 u = up[i];
      w[i].x = fmaf(p, w[i].x, u.x); w[i].y = fmaf(p, w[i].y, u.y);
      w[i].z = fmaf(p, w[i].z, u.z); w[i].w = fmaf(p, w[i].w, u.w);
    }
  }
}

// =====================================================================
// Kernel 3: per (head, chunk):
//   S = Q@Kᵀ ; mask A[i][j] = (j<=i) ? S*(cp_i/cp_j)*(-eta_j) : 0
//   Zq = diag(cp)·(Q@W_base) + A@Grad ; out = ln_fwd(Zq)+q
// =====================================================================
__global__ __launch_bounds__(256) void ttt_k3_out(
    const float* __restrict__ qin, const float* __restrict__ kin,
    const float* __restrict__ grad_ws, const float* __restrict__ cp_ws,
    const float* __restrict__ eta, const float* __restrict__ Wb_ws,
    const float* __restrict__ gamma, const float* __restrict__ beta,
    float* __restrict__ out) {
  __shared__ float Qls[CK * PITCH];
  __shared__ float Kls[CK * PITCH];  // later reused for masked S, then Zq
  __shared__ float Gls[CK * PITCH];
  __shared__ float cp_s[CK];
  __shared__ float eta_s[CK];

  const int blk = blockIdx.x;
  const int h = blk / NC, c = blk % NC;
  const int t0 = c * CK;
  const int tid = threadIdx.x;
  const int lane = tid & 31, wv = tid >> 5;

  const float* Wb = Wb_ws + ((size_t)(h * NC + c)) * D * D;
  if (tid < 64) __builtin_prefetch(Wb + tid * D, 0, 1);  // global_prefetch_b8

  // async-stage Q, K, grad chunks (64x64 each) into LDS
  for (int i = tid; i < CK * D / 4; i += 256) {
    int r = i >> 4, cc = (i & 15) << 2;
    size_t gidx = ((size_t)(h * L + t0 + r)) * D + cc;
    async_copy16(&qin[gidx], &Qls[r * PITCH + cc]);
    async_copy16(&kin[gidx], &Kls[r * PITCH + cc]);
    async_copy16(&grad_ws[gidx], &Gls[r * PITCH + cc]);
  }
  if (tid < CK) {
    cp_s[tid] = cp_ws[h * L + t0 + tid];
    eta_s[tid] = eta[h * L + t0 + tid];
  }
  wait_async0();
  __syncthreads();

  const int ti = wv & 3, tjb = (wv >> 2) << 1;

  // ---- S = Q @ Kᵀ, masked in registers ----
  v8f sacc[2];
#pragma unroll
  for (int tjx = 0; tjx < 2; ++tjx) {
    const int tj = tjb + tjx;
    v8f acc = {};
#pragma unroll
    for (int kb = 0; kb < D / 4; ++kb) {
      v2f a = load_a_rm(Qls, ti * 16, kb * 4, lane);
      v2f b = load_b_tr(Kls, kb * 4, tj * 16, lane);  // B[d][j] = K[j][d]
      acc = wmma_f32_16x16x4(a, b, acc);
    }
    const int j = tj * 16 + (lane & 15);
    const float negeta_cpinv = -eta_s[j] / cp_s[j];
#pragma unroll
    for (int e = 0; e < 8; ++e) {
      int i = ti * 16 + e + ((lane >> 4) << 3);
      float m = (j <= i) ? cp_s[i] * negeta_cpinv : 0.0f;
      acc[e] *= m;
    }
    sacc[tjx] = acc;
  }
  __syncthreads();  // everyone done reading K
#pragma unroll
  for (int tjx = 0; tjx < 2; ++tjx) {
    const int tj = tjb + tjx;
#pragma unroll
    for (int e = 0; e < 8; ++e) {
      int i = ti * 16 + e + ((lane >> 4) << 3);
      Kls[i * PITCH + tj * 16 + (lane & 15)] = sacc[tjx][e];
    }
  }
  __syncthreads();

  // ---- Zq = diag(cp)·(Q @ W_base) + Smask @ Grad ----
  v8f zacc[2];
#pragma unroll
  for (int tjx = 0; tjx < 2; ++tjx) {
    const int tj = tjb + tjx;
    v8f acc = {};
#pragma unroll
    for (int kb = 0; kb < D / 4; ++kb) {
      v2f a = load_a_rm(Qls, ti * 16, kb * 4, lane);
      v2f b = load_b_glb(Wb, kb * 4, tj * 16, lane);
      acc = wmma_f32_16x16x4(a, b, acc);
    }
#pragma unroll
    for (int e = 0; e < 8; ++e) {
      int i = ti * 16 + e + ((lane >> 4) << 3);
      acc[e] *= cp_s[i];
    }
#pragma unroll
    for (int kb = 0; kb < CK / 4; ++kb) {
      v2f a = load_a_rm(Kls, ti * 16, kb * 4, lane);  // masked S
      v2f b = load_b_rm(Gls, kb * 4, tj * 16, lane);
      acc = wmma_f32_16x16x4(a, b, acc);
    }
    zacc[tjx] = acc;
  }
  __syncthreads();  // done reading S/Grad
#pragma unroll
  for (int tjx = 0; tjx < 2; ++tjx) {
    const int tj = tjb + tjx;
#pragma unroll
    for (int e = 0; e < 8; ++e) {
      int i = ti * 16 + e + ((lane >> 4) << 3);
      Kls[i * PITCH + tj * 16 + (lane & 15)] = zacc[tjx][e];  // Zq
    }
  }
  __syncthreads();

  // ---- ln_fwd(Zq) + q ----
  const float g0 = gamma[h * D + lane], g1 = gamma[h * D + lane + 32];
  const float b0 = beta[h * D + lane],  b1 = beta[h * D + lane + 32];
#pragma unroll 1
  for (int rr = 0; rr < 8; ++rr) {
    const int r = wv * 8 + rr;
    float x0 = Kls[r * PITCH + lane], x1 = Kls[r * PITCH + lane + 32];
    float mu = wave_sum32(x0 + x1) * (1.0f / 64.0f);
    float d0 = x0 - mu, d1 = x1 - mu;
    float var = wave_sum32(d0 * d0 + d1 * d1) * (1.0f / 64.0f);
    float rstd = __frsqrt_rn(var + EPSLN);
    float o0 = g0 * (d0 * rstd) + b0 + Qls[r * PITCH + lane];
    float o1 = g1 * (d1 * rstd) + b1 + Qls[r * PITCH + lane + 32];
    out[((size_t)(h * L + t0 + r)) * D + lane] = o0;
    out[((size_t)(h * L + t0 + r)) * D + lane + 32] = o1;
  }
}

// =====================================================================
extern "C" void kernel_launch(void* const* d_in, const int* in_sizes, int n_in,
                              void* d_out, int out_size, void* d_ws,
                              size_t ws_size, hipStream_t stream) {
  const float* q     = (const float*)d_in[0];
  const float* k     = (const float*)d_in[1];
  const float* v     = (const float*)d_in[2];
  const float* gate  = (const float*)d_in[3];
  const float* eta   = (const float*)d_in[4];
  const float* W0    = (const float*)d_in[5];
  const float* gamma = (const float*)d_in[6];
  const float* beta  = (const float*)d_in[7];
  float* out = (float*)d_out;

  float* ws = (float*)d_ws;
  float* grad_ws = ws;                                  // H*L*D
  float* U_ws    = grad_ws + (size_t)H * L * D;         // H*NC*D*D
  float* cp_ws   = U_ws    + (size_t)H * NC * D * D;    // H*L
  float* Pc_ws   = cp_ws   + (size_t)H * L;             // H*NC
  float* Wb_ws   = Pc_ws   + (size_t)H * NC;            // H*NC*D*D

  ttt_k1_grad<<<H * NC, 256, 0, stream>>>(k, v, gate, eta, W0, gamma, beta,
                                          grad_ws, U_ws, cp_ws, Pc_ws);
  ttt_k2_scan<<<H, 256, 0, stream>>>(W0, U_ws, Pc_ws, Wb_ws);
  ttt_k3_out<<<H * NC, 256, 0, stream>>>(q, k, grad_ws, cp_ws, eta, Wb_ws,
                                         gamma, beta, out);
}